// ROI_34230889349163
// MI455X (gfx1250) — compile-verified
//
#include <hip/hip_runtime.h>

typedef __attribute__((ext_vector_type(2))) float v2f;
typedef __attribute__((ext_vector_type(8))) float v8f;

#define HF 50
#define WF 50
#define NCH 256
#define NROI 256
#define OUTS 14
#define KP 52     // K padded 50 -> 52 (13 WMMA k-steps of 4)
#define NP 64     // N padded 50 -> 64 (4 tiles of 16)

// ROI-align as two f32 WMMA GEMMs per channel:
//   P = Wx (16x52) @ F_c (52x64)      [zero-padded]
//   V = P  (16x52) @ Wy^T (52x16)
// then 2x2 max-pool of the valid 14x14 region.
__global__ __launch_bounds__(256) void roi_wmma_kernel(
    const float* __restrict__ proposals,   // (B, N, 4)
    const float* __restrict__ features,    // (B, C, 50, 50)
    float* __restrict__ out)               // (B, N, C, 7, 7)
{
    __shared__ float sWx [16][KP];        // x-interp weights, rows 14..15 zero
    __shared__ float sWyt[KP][16];        // y-interp weights transposed, cols 14..15 zero
    __shared__ float sP  [8][16][NP];     // per-wave intermediate
    __shared__ float sV  [8][16][16];     // per-wave result tile

    const int tid  = threadIdx.x;
    const int wave = tid >> 5;
    const int lane = tid & 31;
    const int half = lane >> 4;           // 0: lanes 0-15, 1: lanes 16-31
    const int l16  = lane & 15;

    const int roi = blockIdx.x;           // b*NROI + n
    const int b   = roi >> 8;

    // ---- zero-init weight matrices (padding must be exactly 0) ----
    for (int idx = tid; idx < 16 * KP; idx += 256) {
        (&sWx[0][0])[idx]  = 0.0f;
        (&sWyt[0][0])[idx] = 0.0f;
    }
    __syncthreads();

    // ---- build interpolation weight matrices ----
    const float fscale = (float)HF / 800.0f;   // Hf/IMG_H == Wf/IMG_W == 0.0625
    const float* prop = proposals + (size_t)roi * 4;
    if (tid < OUTS) {                          // Wx row i: weights along x (dim Hf)
        const int i = tid;
        float x0 = floorf(prop[0] * fscale);
        float w  = ceilf (prop[2] * fscale);
        float t  = (float)i / (float)(OUTS - 1);
        float sx = x0 + t * (w - 1.0f);
        float xf = floorf(sx);
        float ax = sx - xf;
        int x0i = (int)xf;
        x0i = x0i < 0 ? 0 : (x0i > HF - 1 ? HF - 1 : x0i);
        int x1i = x0i + 1 > HF - 1 ? HF - 1 : x0i + 1;
        sWx[i][x0i] += 1.0f - ax;              // same thread: clip-duplicate sums correctly
        sWx[i][x1i] += ax;
    }
    if (tid >= 32 && tid < 32 + OUTS) {        // Wy^T col j: weights along y (dim Wf)
        const int j = tid - 32;
        float y0 = floorf(prop[1] * fscale);
        float h  = ceilf (prop[3] * fscale);
        float t  = (float)j / (float)(OUTS - 1);
        float sy = y0 + t * (h - 1.0f);
        float yf = floorf(sy);
        float ay = sy - yf;
        int y0i = (int)yf;
        y0i = y0i < 0 ? 0 : (y0i > WF - 1 ? WF - 1 : y0i);
        int y1i = y0i + 1 > WF - 1 ? WF - 1 : y0i + 1;
        sWyt[y0i][j] += 1.0f - ay;
        sWyt[y1i][j] += ay;
    }
    __syncthreads();

    const float* featB = features + (size_t)b * NCH * HF * WF;
    float*       outB  = out      + (size_t)roi * NCH * 49;

    for (int it = 0; it < NCH / 8; ++it) {
        const int c = it * 8 + wave;
        const float* F = featB + (size_t)c * HF * WF;

        // ---- stage 1: P = Wx @ F  (4 N-tiles, 13 k-steps each) ----
        for (int nt = 0; nt < 4; ++nt) {
            const int n0  = nt * 16;
            const int col = n0 + l16;
            v8f acc = {0.f, 0.f, 0.f, 0.f, 0.f, 0.f, 0.f, 0.f};
            for (int kk = 0; kk < 13; ++kk) {
                const int k0 = kk * 4;
                const int rA = k0 + 2 * half;
                const int rB = rA + 1;
                v2f a, bm;
                a.x  = sWx[l16][rA];
                a.y  = sWx[l16][rB];
                bm.x = (rA < HF && col < WF) ? F[rA * WF + col] : 0.0f;
                bm.y = (rB < HF && col < WF) ? F[rB * WF + col] : 0.0f;
                acc = __builtin_amdgcn_wmma_f32_16x16x4_f32(
                        false, a, false, bm, (short)0, acc, false, false);
            }
            // C/D layout: VGPR r -> rows r (lanes 0-15) / r+8 (lanes 16-31)
            for (int r = 0; r < 8; ++r)
                sP[wave][r + 8 * half][col] = acc[r];
        }
        __syncthreads();

        // ---- stage 2: V = P @ Wy^T  (1 tile, 13 k-steps) ----
        {
            v8f acc = {0.f, 0.f, 0.f, 0.f, 0.f, 0.f, 0.f, 0.f};
            for (int kk = 0; kk < 13; ++kk) {
                const int k0 = kk * 4;
                const int rA = k0 + 2 * half;
                const int rB = rA + 1;
                v2f a, bm;
                a.x  = sP[wave][l16][rA];
                a.y  = sP[wave][l16][rB];
                bm.x = sWyt[rA][l16];
                bm.y = sWyt[rB][l16];
                acc = __builtin_amdgcn_wmma_f32_16x16x4_f32(
                        false, a, false, bm, (short)0, acc, false, false);
            }
            for (int r = 0; r < 8; ++r)
                sV[wave][r + 8 * half][l16] = acc[r];
        }
        __syncthreads();

        // ---- 2x2 max pool over valid 14x14 and write out ----
        float* oc = outB + (size_t)c * 49;
        for (int idx = lane; idx < 49; idx += 32) {
            const int p = idx / 7, q = idx % 7;
            float m0 = fmaxf(sV[wave][2 * p][2 * q],     sV[wave][2 * p][2 * q + 1]);
            float m1 = fmaxf(sV[wave][2 * p + 1][2 * q], sV[wave][2 * p + 1][2 * q + 1]);
            oc[idx] = fmaxf(m0, m1);
        }
        __syncthreads();
    }
}

extern "C" void kernel_launch(void* const* d_in, const int* in_sizes, int n_in,
                              void* d_out, int out_size, void* d_ws, size_t ws_size,
                              hipStream_t stream) {
    const float* proposals = (const float*)d_in[0];   // (4, 256, 4)
    const float* features  = (const float*)d_in[1];   // (4, 256, 50, 50)
    float* out = (float*)d_out;                       // (4, 256, 256, 7, 7)
    roi_wmma_kernel<<<dim3(4 * NROI), dim3(256), 0, stream>>>(proposals, features, out);
}